// InstanceConsistencyLoss_44710609551555
// MI455X (gfx1250) — compile-verified
//
#include <hip/hip_runtime.h>

#define B_   8
#define C_   128
#define H_   512
#define W_   512
#define P_   (H_*W_)       // 262144 pixels per plane
#define S_   257           // segments per image (0 = background)
#define G_   (B_*S_)       // 2056 total segments

typedef __attribute__((ext_vector_type(2))) float v2f;
typedef __attribute__((ext_vector_type(8))) float v8f;

// ---------------------------------------------------------------- zero scratch
__global__ void icl_zero_ws(unsigned* __restrict__ ws, int n) {
  int i = blockIdx.x * blockDim.x + threadIdx.x;
  int stride = gridDim.x * blockDim.x;
  for (; i < n; i += stride) ws[i] = 0u;
}

// ---------------------------------------------------------------- pixel counts
// 64 blocks: b = blk>>3, chunk = blk&7 ; each block covers 32768 pixels.
__global__ void icl_count(const int* __restrict__ ids, unsigned* __restrict__ cnt) {
  __shared__ unsigned lcnt[S_];
  for (int s = threadIdx.x; s < S_; s += blockDim.x) lcnt[s] = 0u;
  __syncthreads();

  int blk = blockIdx.x;
  int b = blk >> 3, chunk = blk & 7;
  const int4* i4 = reinterpret_cast<const int4*>(ids + (size_t)b * P_ + (size_t)chunk * 32768);
  int lane = threadIdx.x & 31;
  int gb = lane & ~7;                    // 8-lane group base (32 aligned pixels)

  for (int it = 0; it < 32; ++it) {
    int4 idv = i4[it * 256 + threadIdx.x];
    bool ok4 = (idv.x == idv.y) && (idv.x == idv.z) && (idv.x == idv.w);
    int id0 = __shfl(idv.x, gb, 32);
    unsigned bal = (unsigned)__ballot(ok4 && (idv.x == id0));
    bool grpUni = ((bal >> gb) & 0xFFu) == 0xFFu;   // group-uniform over 32 pixels
    if (grpUni) {
      if ((lane & 7) == 0) atomicAdd(&lcnt[idv.x], 32u);
    } else if (ok4) {
      atomicAdd(&lcnt[idv.x], 4u);
    } else {                                        // exact general fallback
      atomicAdd(&lcnt[idv.x], 1u); atomicAdd(&lcnt[idv.y], 1u);
      atomicAdd(&lcnt[idv.z], 1u); atomicAdd(&lcnt[idv.w], 1u);
    }
  }
  __syncthreads();
  for (int s = threadIdx.x; s < S_; s += blockDim.x) {
    unsigned v = lcnt[s];
    if (v) atomicAdd(&cnt[b * S_ + s], v);
  }
}

// ------------------------------------------------------- segment sum / sum-sq
// 8192 blocks: plane = blk>>3 (= b*C + c), chunk = blk&7; 32768 pixels/block.
__global__ void icl_accum(const float* __restrict__ feat, const int* __restrict__ ids,
                          float* __restrict__ Sum, float* __restrict__ Sum2) {
  __shared__ float lsum[S_];
  __shared__ float lsum2[S_];
  for (int s = threadIdx.x; s < S_; s += blockDim.x) { lsum[s] = 0.f; lsum2[s] = 0.f; }
  __syncthreads();

  int blk = blockIdx.x;
  int chunk = blk & 7;
  int plane = blk >> 3;          // b*C + c
  int b = plane >> 7;            // / 128
  int c = plane & 127;
  const float4* f4 = reinterpret_cast<const float4*>(feat + (size_t)plane * P_ + (size_t)chunk * 32768);
  const int4*  i4 = reinterpret_cast<const int4*>(ids + (size_t)b * P_ + (size_t)chunk * 32768);
  int lane = threadIdx.x & 31;
  int gb = lane & ~7;

  for (int it = 0; it < 32; ++it) {
    float4 f = f4[it * 256 + threadIdx.x];
    int4 idv = i4[it * 256 + threadIdx.x];
    float s1 = (f.x + f.y) + (f.z + f.w);
    float s2 = fmaf(f.x, f.x, fmaf(f.y, f.y, fmaf(f.z, f.z, f.w * f.w)));
    bool ok4 = (idv.x == idv.y) && (idv.x == idv.z) && (idv.x == idv.w);
    int id0 = __shfl(idv.x, gb, 32);
    unsigned bal = (unsigned)__ballot(ok4 && (idv.x == id0));
    bool grpUni = ((bal >> gb) & 0xFFu) == 0xFFu;
    if (grpUni) {               // 32 consecutive pixels share one id: shuffle tree
      s1 += __shfl_xor(s1, 1, 32); s2 += __shfl_xor(s2, 1, 32);
      s1 += __shfl_xor(s1, 2, 32); s2 += __shfl_xor(s2, 2, 32);
      s1 += __shfl_xor(s1, 4, 32); s2 += __shfl_xor(s2, 4, 32);
      if ((lane & 7) == 0) { atomicAdd(&lsum[idv.x], s1); atomicAdd(&lsum2[idv.x], s2); }
    } else if (ok4) {
      atomicAdd(&lsum[idv.x], s1); atomicAdd(&lsum2[idv.x], s2);
    } else {                    // exact general fallback
      atomicAdd(&lsum[idv.x], f.x); atomicAdd(&lsum2[idv.x], f.x * f.x);
      atomicAdd(&lsum[idv.y], f.y); atomicAdd(&lsum2[idv.y], f.y * f.y);
      atomicAdd(&lsum[idv.z], f.z); atomicAdd(&lsum2[idv.z], f.z * f.z);
      atomicAdd(&lsum[idv.w], f.w); atomicAdd(&lsum2[idv.w], f.w * f.w);
    }
  }
  __syncthreads();
  for (int s = threadIdx.x; s < S_; s += blockDim.x) {
    float v1 = lsum[s], v2 = lsum2[s];
    if (v2 != 0.f) {            // skipping exact zeros is identity-safe
      size_t o = (size_t)(b * S_ + s) * C_ + c;
      atomicAdd(&Sum[o], v1);
      atomicAdd(&Sum2[o], v2);
    }
  }
}

// ------------------------------------- per-segment variance via f32 WMMA rowsum
// One wave per 16-segment tile. D(16x16) += A(16x4) x Ones(4x16) over 32 k-steps
// reduces 128 channels exactly in f32 (V_WMMA_F32_16X16X4_F32, RNE).
__global__ void icl_var_reduce(const float* __restrict__ Sum, const float* __restrict__ Sum2,
                               const unsigned* __restrict__ cnt, float* __restrict__ img) {
  int tile = blockIdx.x;
  int lane = threadIdx.x;              // blockDim.x == 32, EXEC all-1s
  int m  = lane & 15;
  int kk = (lane >> 4) << 1;           // lanes 0-15 -> K {0,1}; lanes 16-31 -> K {2,3}
  int seg = tile * 16 + m;
  int segc = seg < G_ ? seg : (G_ - 1);
  bool segok = seg < G_;
  const float* rS  = Sum  + (size_t)segc * C_;
  const float* rS2 = Sum2 + (size_t)segc * C_;

  v8f accA = {0.f, 0.f, 0.f, 0.f, 0.f, 0.f, 0.f, 0.f};   // rowsum of Sum2
  v8f accB = {0.f, 0.f, 0.f, 0.f, 0.f, 0.f, 0.f, 0.f};   // rowsum of Sum^2
  v2f ones; ones.x = 1.f; ones.y = 1.f;

  for (int kb = 0; kb < C_; kb += 4) {
    float a0 = rS2[kb + kk], a1 = rS2[kb + kk + 1];
    float b0 = rS [kb + kk], b1 = rS [kb + kk + 1];
    a0 = segok ? a0 : 0.f;  a1 = segok ? a1 : 0.f;        // select, no EXEC change
    b0 = segok ? b0 : 0.f;  b1 = segok ? b1 : 0.f;
    v2f A;  A.x = a0;      A.y = a1;
    v2f Bq; Bq.x = b0*b0;  Bq.y = b1*b1;
    accA = __builtin_amdgcn_wmma_f32_16x16x4_f32(false, A,  false, ones, (short)0, accA, false, false);
    accB = __builtin_amdgcn_wmma_f32_16x16x4_f32(false, Bq, false, ones, (short)0, accB, false, false);
  }

  // D layout: lane 0 holds rows 0..7 (VGPR 0..7, N=0); lane 16 holds rows 8..15.
  if (lane == 0 || lane == 16) {
    int basem = tile * 16 + (lane == 16 ? 8 : 0);
#pragma unroll
    for (int v = 0; v < 8; ++v) {
      int sg = basem + v;
      if (sg < G_) {
        int bi = sg / S_;
        int sl = sg - bi * S_;
        unsigned cn = cnt[sg];
        if (sl != 0 && cn != 0u) {       // drop background & empty segments
          float cf = (float)cn;
          float V = (accA[v] - accB[v] / cf) / cf;
          atomicAdd(&img[bi], V);
          atomicAdd(&img[8 + bi], 1.0f);
        }
      }
    }
  }
}

// ---------------------------------------------------------------- final scalar
__global__ void icl_final(const float* __restrict__ img, float* __restrict__ out) {
  if (threadIdx.x == 0 && blockIdx.x == 0) {
    float acc = 0.f;
    for (int b = 0; b < B_; ++b) {
      float n = img[8 + b];
      acc += (n > 0.f) ? (img[b] / n) : 0.f;
    }
    out[0] = acc / (float)B_;
  }
}

extern "C" void kernel_launch(void* const* d_in, const int* in_sizes, int n_in,
                              void* d_out, int out_size, void* d_ws, size_t ws_size,
                              hipStream_t stream) {
  const float* feat = (const float*)d_in[0];   // (B, C, H, W) float32
  const int*   ids  = (const int*)d_in[1];     // (B, H, W) int32

  // Workspace layout (floats / ints, contiguous):
  //   Sum  : G_*C_ floats
  //   Sum2 : G_*C_ floats
  //   img  : 16 floats  (Vsum[8], n_inst[8])
  //   cnt  : G_ uint32
  float*    Sum  = (float*)d_ws;
  float*    Sum2 = Sum + (size_t)G_ * C_;
  float*    img  = Sum2 + (size_t)G_ * C_;
  unsigned* cnt  = (unsigned*)(img + 16);
  int ndwords = G_ * C_ * 2 + 16 + G_;

  icl_zero_ws<<<256, 256, 0, stream>>>((unsigned*)d_ws, ndwords);
  icl_count<<<B_ * 8, 256, 0, stream>>>(ids, cnt);
  icl_accum<<<B_ * C_ * 8, 256, 0, stream>>>(feat, ids, Sum, Sum2);
  icl_var_reduce<<<(G_ + 15) / 16, 32, 0, stream>>>(Sum, Sum2, cnt, img);
  icl_final<<<1, 32, 0, stream>>>(img, (float*)d_out);
}